// RecurrentSNN_73615739453504
// MI455X (gfx1250) — compile-verified
//
#include <hip/hip_runtime.h>
#include <math.h>

// ---------------- problem constants ----------------
#define NN 6144
#define INPUT_SIZE 1024
#define OUTPUT_SIZE 10
#define DT 1.0f
#define TAU_MEM 10.0f
#define V_THRESH 1.0f
#define V_RESET 0.0f
#define BETA 0.95f
#define A_PLUS 0.008f
#define A_MINUS (-0.0084f)
#define TAU_PLUS 16.8f
#define TAU_MINUS 33.7f
#define MAX_WEIGHT 1.5f
#define WEIGHT_DECAY 1e-7f

// ---------------- flat output offsets (return-tuple order) ----------------
#define OFF_SPK  0                         // current_spikes  (6144)
#define OFF_ACT  6144                      // action_rates    (10)
#define OFF_V    6154                      // V_new           (6144)
#define OFF_MEM  12298                     // mem_new         (6144)
#define OFF_WREC 18442                     // W_rec_new       (6144*6144)
#define OFF_HIST 37767178ull               // spike_history   (6144*2)

#define KSPLIT 8
#define KCHUNK (NN / KSPLIT)               // 768 rows per split-K wave

typedef float v2f __attribute__((ext_vector_type(2)));
typedef float v8f __attribute__((ext_vector_type(8)));

// =====================================================================
// Kernel 1: rec_input = pre @ W_rec via V_WMMA_F32_16X16X4_F32.
// One wave per (16-column tile, K-split segment). A = pre broadcast over M,
// B = 4x16 tile of W_rec. Zero K-chunks (all 4 pre spikes == 0) are skipped
// with an explicitly SCALAR branch (readfirstlane) so EXEC is provably
// all-ones at every WMMA and control flow stays on the SALU.
// =====================================================================
__global__ void __launch_bounds__(256)
rec_gemv_wmma(const float* __restrict__ hist,     // (N,2): pre[i] = hist[2i+1]
              const float* __restrict__ W_rec,    // (N,N) row-major
              float* __restrict__ partial) {      // (KSPLIT, N)
    const int lane = threadIdx.x & 31;
    const int wave = threadIdx.x >> 5;            // split-K id 0..7
    const int col0 = blockIdx.x * 16;             // column tile
    const int half = lane >> 4;                   // 0: K=0,1  1: K=2,3
    const int nn   = lane & 15;
    const int kbeg = wave * KCHUNK;
    const int kend = kbeg + KCHUNK;

    float rec = 0.0f;
#if __has_builtin(__builtin_amdgcn_wmma_f32_16x16x4_f32)
    v8f acc = {};
    for (int k0 = kbeg; k0 < kend; k0 += 4) {
        // all lanes read the same addresses -> values wave-uniform
        float p0 = hist[(k0 + 0) * 2 + 1];
        float p1 = hist[(k0 + 1) * 2 + 1];
        float p2 = hist[(k0 + 2) * 2 + 1];
        float p3 = hist[(k0 + 3) * 2 + 1];
        // spikes are {0.0,1.0}: sum bit-pattern is 0 iff all four are zero.
        // readfirstlane makes the branch scalar (s_cmp/s_cbranch, EXEC intact).
        unsigned szero =
            __builtin_amdgcn_readfirstlane(__float_as_uint(p0 + p1 + p2 + p3));
        if (szero == 0u) continue;

        const int kk = k0 + half * 2;
        v2f a, b;
        a.x = half ? p2 : p0;                     // A[m][k] = pre[k] (all m)
        a.y = half ? p3 : p1;
        b.x = W_rec[kk * NN + col0 + nn];         // B[k][n] = W_rec[k][col0+n]
        b.y = W_rec[(kk + 1) * NN + col0 + nn];
        acc = __builtin_amdgcn_wmma_f32_16x16x4_f32(
                  false, a, false, b, (short)0, acc, false, false);
    }
    rec = acc[0];                                 // M=0 row; valid on lanes 0..15
#else
    if (lane < 16) {
        for (int k = kbeg; k < kend; ++k) {
            float p = hist[k * 2 + 1];
            if (p != 0.0f) rec += p * W_rec[k * NN + col0 + lane];
        }
    }
#endif
    if (lane < 16) partial[wave * NN + col0 + nn] = rec;
}

// =====================================================================
// Kernel 2: split-K reduce + input GEMV (skip zero input rows) + LIF update
// =====================================================================
__global__ void __launch_bounds__(256)
neuron_update(const float* __restrict__ input_spk,
              const float* __restrict__ V,
              const float* __restrict__ hist,
              const float* __restrict__ mem,
              const float* __restrict__ W_in,     // (INPUT_SIZE, N)
              const float* __restrict__ noise,
              const float* __restrict__ partial,
              float* __restrict__ out) {
    const int j = blockIdx.x * blockDim.x + threadIdx.x;
    if (j >= NN) return;

    float cur = 0.0f;
#pragma unroll
    for (int s = 0; s < KSPLIT; ++s) cur += partial[s * NN + j];

    for (int k = 0; k < INPUT_SIZE; ++k) {        // ~10% dense
        float s = input_spk[k];
        unsigned sz = __builtin_amdgcn_readfirstlane(__float_as_uint(s));
        if (sz != 0u) cur += s * W_in[k * NN + j];   // scalar skip branch
    }

    const float decay = __expf(-DT / TAU_MEM);
    float v = V[j] * decay + cur + 0.015f * noise[j];
    const bool fired = (v >= V_THRESH);
    const float spike = fired ? 1.0f : 0.0f;
    v = fired ? V_RESET : v;
    const float m = BETA * mem[j] + (1.0f - BETA) * spike;

    out[OFF_SPK + j] = spike;
    out[OFF_V + j]   = v;
    out[OFF_MEM + j] = m;
    out[OFF_HIST + 2 * (size_t)j]     = hist[2 * j + 1];  // shift history
    out[OFF_HIST + 2 * (size_t)j + 1] = spike;
}

// =====================================================================
// Kernel 3: STDP weight update — the 302 MB stream (read W, write W_new).
// float4 loads; non-temporal 8-byte vector stores (OFF_WREC is only
// 8B-aligned in d_out, and the output stream must not evict W_rec from
// the 192 MB L2 before this kernel finishes re-reading it).
// =====================================================================
__global__ void __launch_bounds__(256)
stdp_update(const float* __restrict__ W_rec,
            const float* __restrict__ hist,
            float* out) {
    const float kp = A_PLUS  * __expf(-DT / TAU_PLUS);
    const float km = A_MINUS * __expf( DT / TAU_MINUS);

    const int idx = blockIdx.x * blockDim.x + threadIdx.x;  // N*N/4 threads
    const int i   = idx / (NN / 4);
    const int j0  = (idx - i * (NN / 4)) * 4;

    const float pre_i  = hist[2 * i + 1];
    const float post_i = out[OFF_SPK + i];

    const float4 w4 = *(const float4*)(W_rec + (size_t)i * NN + j0);
    const float4 p4 = *(const float4*)(out + OFF_SPK + j0);     // post[j0..3]
    const float4 h0 = *(const float4*)(hist + 2 * j0);          // pre[j0..1]
    const float4 h1 = *(const float4*)(hist + 2 * j0 + 4);      // pre[j2..3]

    float w[4]      = {w4.x, w4.y, w4.z, w4.w};
    float post_j[4] = {p4.x, p4.y, p4.z, p4.w};
    float pre_j[4]  = {h0.y, h0.w, h1.y, h1.w};

    float nw[4];
#pragma unroll
    for (int e = 0; e < 4; ++e) {
        float dw = kp * pre_i * post_j[e] + km * post_i * pre_j[e];
        float wv = w[e] + ((w[e] != 0.0f) ? dw : 0.0f);
        wv *= (1.0f - WEIGHT_DECAY);
        nw[e] = fminf(MAX_WEIGHT, fmaxf(-MAX_WEIGHT, wv));
    }

    v2f* dst = (v2f*)(out + OFF_WREC + (size_t)i * NN + j0);
    v2f lo, hi;
    lo.x = nw[0]; lo.y = nw[1];
    hi.x = nw[2]; hi.y = nw[3];
    __builtin_nontemporal_store(lo, dst);
    __builtin_nontemporal_store(hi, dst + 1);
}

// =====================================================================
// Kernel 4: action_rates = spikes @ W_out — 10 waves, one per output.
// =====================================================================
__global__ void __launch_bounds__(320)
readout(const float* __restrict__ W_out, float* out) {
    const int o    = threadIdx.x >> 5;            // 0..9
    const int lane = threadIdx.x & 31;
    float sum = 0.0f;
    for (int j = lane; j < NN; j += 32) {
        float s = out[OFF_SPK + j];
        if (s != 0.0f) sum += s * W_out[j * OUTPUT_SIZE + o];
    }
#pragma unroll
    for (int off = 16; off > 0; off >>= 1)
        sum += __shfl_down(sum, off, 32);
    if (lane == 0) out[OFF_ACT + o] = sum;
}

// =====================================================================
extern "C" void kernel_launch(void* const* d_in, const int* in_sizes, int n_in,
                              void* d_out, int out_size, void* d_ws, size_t ws_size,
                              hipStream_t stream) {
    const float* input_spk = (const float*)d_in[0];   // (1,1024)
    const float* V         = (const float*)d_in[1];   // (1,N)
    const float* hist      = (const float*)d_in[2];   // (1,N,2)
    const float* mem       = (const float*)d_in[3];   // (1,N)
    const float* W_rec     = (const float*)d_in[4];   // (N,N)
    const float* W_in      = (const float*)d_in[5];   // (1024,N)
    const float* W_out     = (const float*)d_in[6];   // (N,10)
    const float* noise     = (const float*)d_in[7];   // (1,N)
    float* out = (float*)d_out;
    float* partial = (float*)d_ws;                    // KSPLIT*N floats

    // 1) recurrent GEMV (WMMA, split-K, every partial slot written)
    rec_gemv_wmma<<<NN / 16, 256, 0, stream>>>(hist, W_rec, partial);
    // 2) neuron update (consumes partials, produces spikes/V/mem/history)
    neuron_update<<<(NN + 255) / 256, 256, 0, stream>>>(
        input_spk, V, hist, mem, W_in, noise, partial, out);
    // 3) STDP stream over W_rec (dominant ~302 MB of HBM traffic)
    stdp_update<<<(NN / 4) * (NN / 256), 256, 0, stream>>>(W_rec, hist, out);
    // 4) readout
    readout<<<1, 320, 0, stream>>>(W_out, out);
}